// BaseThinker_38766374814195
// MI455X (gfx1250) — compile-verified
//
#include <hip/hip_runtime.h>
#include <math.h>

// Problem constants (from reference)
#define N_MEMS   1000000
#define MEM_DIM  88
#define OBS_LEN  64
#define ACT_LEN  16
#define RET_LEN  8
#define BQ       64
#define TOPK     16

// Tiling
#define NBLK     512      // blocks in main kernel
#define TILE     64       // memories per pass per block
#define OPAD     68       // padded LDS row stride (floats): 272B rows, 16B aligned, bank-spread

#define FINF  (__builtin_inff())

typedef __attribute__((ext_vector_type(2))) float v2f;
typedef __attribute__((ext_vector_type(8))) float v8f;

// ---------------------------------------------------------------------------
// CDNA5 async global -> LDS copy (per-lane 16B), tracked by ASYNCcnt.
// lds_off: byte offset in LDS (low 32 bits of a generic __shared__ pointer).
// ---------------------------------------------------------------------------
__device__ __forceinline__ void async_copy_b128(unsigned lds_off, const float* gptr) {
    asm volatile("global_load_async_to_lds_b128 %0, %1, off"
                 :
                 : "v"(lds_off), "v"((unsigned long long)(uintptr_t)gptr)
                 : "memory");
}
__device__ __forceinline__ void wait_asynccnt0() {
    asm volatile("s_wait_asynccnt 0x0" ::: "memory");
}

// ---------------------------------------------------------------------------
// Sorted ascending top-16 insertion. Fully unrolled; all array indices are
// compile-time constants so lists live in VGPRs.
// ---------------------------------------------------------------------------
__device__ __forceinline__ void insert16(float s, int id, float* ls, int* li) {
    if (s >= ls[15]) return;   // common fast path
#pragma unroll
    for (int j = 15; j >= 1; --j) {
        float o1 = ls[j - 1];
        float o0 = (j >= 2) ? ls[j - 2] : -FINF;
        if (o1 > s) {
            ls[j] = o1;
            li[j] = li[j - 1];
            if (o0 <= s) { ls[j - 1] = s; li[j - 1] = id; }
        }
    }
}

// ---------------------------------------------------------------------------
// Kernel 1: normalize obs rows (64 x 64). One block of 64 threads.
// ---------------------------------------------------------------------------
__global__ void obs_normalize_kernel(const float* __restrict__ obs,
                                     float* __restrict__ obs_n) {
    int r = threadIdx.x;   // 0..63
    float ss = 0.f;
#pragma unroll 8
    for (int j = 0; j < OBS_LEN; ++j) {
        float v = obs[r * OBS_LEN + j];
        ss += v * v;
    }
    float nrm = fmaxf(sqrtf(ss), 1e-12f);
    float inv = 1.0f / nrm;
#pragma unroll 8
    for (int j = 0; j < OBS_LEN; ++j)
        obs_n[r * OBS_LEN + j] = obs[r * OBS_LEN + j] * inv;
}

// ---------------------------------------------------------------------------
// One 16x16 WMMA tile task: dot(obs_n[Mtile], memT[Ctile]) over K=64,
// then score = ||mem||^2 - 2*dot written to the LDS score tile.
// Straight-line, no branches: EXEC stays all-ones across the WMMAs.
// ---------------------------------------------------------------------------
__device__ __forceinline__ void wmma_tile_task(int task, int lane,
                                               const float* OBS, const float* MEMT,
                                               const float* NRM, float* SCR,
                                               int mBase, int mEnd) {
    const int Mtile = task >> 2;
    const int Ctile = task & 3;
    const int arow  = Mtile * 16 + (lane & 15);
    const int brow  = Ctile * 16 + (lane & 15);
    const int kb    = (lane >= 16) ? 2 : 0;   // f32 16x16x4: hi lanes hold K=+2/+3

    v8f c = {};
#pragma unroll
    for (int ks = 0; ks < 16; ++ks) {
        v2f a = *(const v2f*)(OBS  + arow * OPAD + ks * 4 + kb);
        v2f b = *(const v2f*)(MEMT + brow * OPAD + ks * 4 + kb);
        c = __builtin_amdgcn_wmma_f32_16x16x4_f32(
                false, a, false, b, (short)0, c, false, false);
    }

    // C/D layout: VGPR r, lanes 0-15 -> M = r, lanes 16-31 -> M = r+8.
    const int   col   = Ctile * 16 + (lane & 15);
    const int   rbase = Mtile * 16 + ((lane >= 16) ? 8 : 0);
    const float nc    = NRM[col];
    const bool  valid = (mBase + col) < mEnd;
#pragma unroll
    for (int r = 0; r < 8; ++r) {
        float s = valid ? fmaf(-2.0f, c[r], nc) : FINF;
        SCR[(rbase + r) * 64 + col] = s;
    }
}

// ---------------------------------------------------------------------------
// Kernel 2: fused WMMA distance + per-block top-16 per query row.
// ---------------------------------------------------------------------------
__global__ void __launch_bounds__(256)
dist_topk_kernel(const float* __restrict__ mem,
                 const float* __restrict__ obs_n,
                 float* __restrict__ cand_s,
                 int*   __restrict__ cand_i) {
    // LDS pool, manually carved & partially reused:
    //   [0      , 17408) obs tile   [64][OPAD]
    //   [17408  , 34816) mem tile   [64][OPAD]   (reused as CAND scores)
    //   [34816  , 35072) norms      [64]
    //   [35072  , 51456) score tile [64][64]     (reused as CAND indices)
    __shared__ __align__(16) unsigned char smem[51456];
    float* OBS  = (float*)(smem);
    float* MEMT = (float*)(smem + 17408);
    float* NRM  = (float*)(smem + 34816);
    float* SCR  = (float*)(smem + 35072);
    float* CS   = (float*)(smem + 17408);   // 256*16 floats (reuse MEMT)
    int*   CI   = (int*)  (smem + 33792);   // 256*16 ints   (reuse tail+SCR)

    const unsigned memt_lds = (unsigned)(uintptr_t)(void*)MEMT;  // LDS byte offset

    const int t    = threadIdx.x;
    const int lane = t & 31;
    const int wave = t >> 5;
    const int blk  = blockIdx.x;

    const int per    = (N_MEMS + NBLK - 1) / NBLK;      // 1954
    const int mStart = blk * per;
    const int mEnd   = (mStart + per < N_MEMS) ? (mStart + per) : N_MEMS;
    const int nPass  = (per + TILE - 1) / TILE;         // 31 (uniform)

    // Load normalized obs (64x64) into LDS once, coalesced float4.
#pragma unroll
    for (int j = 0; j < 4; ++j) {
        int li  = t + 256 * j;           // 0..1023
        int row = li >> 4, c4 = li & 15;
        float4 v = ((const float4*)(obs_n + row * OBS_LEN))[c4];
        ((float4*)(OBS + row * OPAD))[c4] = v;
    }

    // Register-resident running top-16 for (row = t>>2, col quarter = t&3).
    float ls[TOPK];
    int   lidx[TOPK];
#pragma unroll
    for (int j = 0; j < TOPK; ++j) { ls[j] = FINF; lidx[j] = 0; }
    const int myrow = t >> 2;
    const int myq   = t & 3;

    for (int p = 0; p < nPass; ++p) {
        const int mBase = mStart + p * TILE;

        // Stage 64 memory rows x 64 dims straight into LDS with CDNA5 async
        // copies (rows are 352B, so the 256B obs-prefix is 16B aligned).
        // Out-of-range rows are clamped (their columns are forced to +INF
        // later, so their contents never reach the top-k).
#pragma unroll
        for (int j = 0; j < 4; ++j) {
            int li  = t + 256 * j;
            int row = li >> 4, c4 = li & 15;
            int g   = mBase + row;
            if (g >= N_MEMS) g = N_MEMS - 1;
            async_copy_b128(memt_lds + (unsigned)(row * OPAD + c4 * 4) * 4u,
                            mem + (size_t)g * MEM_DIM + c4 * 4);
        }
        wait_asynccnt0();
        __syncthreads();

        // Per-memory squared norm of the obs-part (float4 reads, aligned).
        if (t < TILE) {
            float ss = 0.f;
#pragma unroll
            for (int j = 0; j < 16; ++j) {
                float4 v = ((const float4*)(MEMT + t * OPAD))[j];
                ss += v.x * v.x + v.y * v.y + v.z * v.z + v.w * v.w;
            }
            NRM[t] = ss;
        }
        __syncthreads();

        // Exactly two tile tasks per wave, straight-line (no exec-masked loop).
        wmma_tile_task(wave,     lane, OBS, MEMT, NRM, SCR, mBase, mEnd);
        wmma_tile_task(wave + 8, lane, OBS, MEMT, NRM, SCR, mBase, mEnd);
        __syncthreads();

        // Each thread scans its 16 columns of its row, updates local top-16.
#pragma unroll
        for (int j = 0; j < 16; ++j) {
            int col = myq * 16 + j;
            insert16(SCR[myrow * 64 + col], mBase + col, ls, lidx);
        }
        __syncthreads();   // protect MEMT/SCR before next pass overwrites
    }

    // Dump per-thread lists to LDS (reused region), then 1 thread/row merges 4.
#pragma unroll
    for (int j = 0; j < TOPK; ++j) { CS[t * 16 + j] = ls[j]; CI[t * 16 + j] = lidx[j]; }
    __syncthreads();

    if (myq == 0) {
        float ms[TOPK];
        int   mi[TOPK];
#pragma unroll
        for (int j = 0; j < TOPK; ++j) { ms[j] = FINF; mi[j] = 0; }
        for (int u = 0; u < 4; ++u)
#pragma unroll
            for (int j = 0; j < 16; ++j)
                insert16(CS[(t + u) * 16 + j], CI[(t + u) * 16 + j], ms, mi);
        size_t base = ((size_t)blk * BQ + myrow) * TOPK;
#pragma unroll
        for (int j = 0; j < TOPK; ++j) { cand_s[base + j] = ms[j]; cand_i[base + j] = mi[j]; }
    }
}

// ---------------------------------------------------------------------------
// Kernel 3: per query row, merge NBLK*16 candidates -> global top-16,
// pick the hit with max return-sum, emit its 16-float action.
// ---------------------------------------------------------------------------
__global__ void __launch_bounds__(256)
final_select_kernel(const float* __restrict__ mem,
                    const float* __restrict__ cand_s,
                    const int*   __restrict__ cand_i,
                    float* __restrict__ out) {
    __shared__ float CS[256 * 16];
    __shared__ int   CI[256 * 16];
    __shared__ float S2[16 * 16];
    __shared__ int   I2[16 * 16];
    __shared__ int   FI[16];
    __shared__ float RS[16];
    __shared__ int   bestIdx;

    const int row = blockIdx.x;
    const int t   = threadIdx.x;

    float ls[TOPK];
    int   li[TOPK];
#pragma unroll
    for (int j = 0; j < TOPK; ++j) { ls[j] = FINF; li[j] = 0; }

    // 8192 candidates for this row; each thread scans 32 (coalesced).
    for (int u = 0; u < (NBLK * TOPK) / 256; ++u) {
        int    e    = t + 256 * u;
        int    blk  = e >> 4, j = e & 15;
        size_t base = ((size_t)blk * BQ + row) * TOPK + j;
        insert16(cand_s[base], cand_i[base], ls, li);
    }
#pragma unroll
    for (int j = 0; j < TOPK; ++j) { CS[t * 16 + j] = ls[j]; CI[t * 16 + j] = li[j]; }
    __syncthreads();

    // Tree merge: 16 threads x 16 lists -> 16 lists, then thread 0 -> final 16.
    if (t < 16) {
        float ms[TOPK];
        int   mi[TOPK];
#pragma unroll
        for (int j = 0; j < TOPK; ++j) { ms[j] = FINF; mi[j] = 0; }
        for (int u = 0; u < 16; ++u)
#pragma unroll
            for (int j = 0; j < 16; ++j)
                insert16(CS[(t * 16 + u) * 16 + j], CI[(t * 16 + u) * 16 + j], ms, mi);
#pragma unroll
        for (int j = 0; j < TOPK; ++j) { S2[t * 16 + j] = ms[j]; I2[t * 16 + j] = mi[j]; }
    }
    __syncthreads();

    if (t == 0) {
        float ms[TOPK];
        int   mi[TOPK];
#pragma unroll
        for (int j = 0; j < TOPK; ++j) { ms[j] = FINF; mi[j] = 0; }
        for (int e = 0; e < 256; ++e)
            insert16(S2[e], I2[e], ms, mi);
#pragma unroll
        for (int j = 0; j < TOPK; ++j) FI[j] = mi[j];
    }
    __syncthreads();

    // Return-sums of the 16 nearest hits.
    if (t < 16) {
        int   idx = FI[t];
        float rs  = 0.f;
#pragma unroll
        for (int q = 0; q < RET_LEN; ++q)
            rs += mem[(size_t)idx * MEM_DIM + OBS_LEN + ACT_LEN + q];
        RS[t] = rs;
    }
    __syncthreads();

    if (t == 0) {
        float best = -FINF;
        int   bi   = FI[0];
        for (int j = 0; j < 16; ++j)
            if (RS[j] > best) { best = RS[j]; bi = FI[j]; }   // first max, like argmax
        bestIdx = bi;
    }
    __syncthreads();

    if (t < ACT_LEN)
        out[row * ACT_LEN + t] = mem[(size_t)bestIdx * MEM_DIM + OBS_LEN + t];
}

// ---------------------------------------------------------------------------
// Host-side launcher
// ---------------------------------------------------------------------------
extern "C" void kernel_launch(void* const* d_in, const int* in_sizes, int n_in,
                              void* d_out, int out_size, void* d_ws, size_t ws_size,
                              hipStream_t stream) {
    (void)in_sizes; (void)n_in; (void)out_size; (void)ws_size;

    const float* obs = (const float*)d_in[0];   // 64*64 f32
    const float* mem = (const float*)d_in[1];   // 1e6*88 f32
    float*       out = (float*)d_out;           // 64*16 f32

    // Workspace layout (floats): obs_n[4096] | cand_s[NBLK*64*16] | cand_i[...]
    float* obs_n  = (float*)d_ws;
    float* cand_s = obs_n + BQ * OBS_LEN;
    int*   cand_i = (int*)(cand_s + (size_t)NBLK * BQ * TOPK);

    obs_normalize_kernel<<<1, 64, 0, stream>>>(obs, obs_n);
    dist_topk_kernel<<<NBLK, 256, 0, stream>>>(mem, obs_n, cand_s, cand_i);
    final_select_kernel<<<BQ, 256, 0, stream>>>(mem, cand_s, cand_i, out);
}